// COOLCHIC_INTERP_ENC_7017976562282
// MI455X (gfx1250) — compile-verified
//
#include <hip/hip_runtime.h>
#include <hip/hip_bf16.h>
#include <stdint.h>

#define H_GRID 721
#define W_GRID 1440
#define NLEVEL 8

typedef unsigned int u32x4 __attribute__((ext_vector_type(4)));
typedef int          i32x8 __attribute__((ext_vector_type(8)));
typedef int          i32x4 __attribute__((ext_vector_type(4)));
typedef float        f32x2 __attribute__((ext_vector_type(2)));
typedef float        f32x4 __attribute__((ext_vector_type(4)));

static __device__ __forceinline__ int imin(int a, int b) { return a < b ? a : b; }
static __device__ __forceinline__ int imax(int a, int b) { return a > b ? a : b; }

// Levels 4..7 touch only a tiny corner of the (same-shaped) 721x1440 grid:
//   rows_l = (720>>l)+2, cols_l = (1440>>l)+2  (covers clamped floor+1 indices)
// Staged tiles are packed contiguously in LDS with pitch = cols_l.
#define SROWS4 47
#define SCOLS4 92
#define SROWS5 24
#define SCOLS5 47
#define SROWS6 13
#define SCOLS6 24
#define SROWS7 7
#define SCOLS7 13
#define SOFF4 0
#define SOFF5 (SOFF4 + SROWS4 * SCOLS4)   /* 4324 */
#define SOFF6 (SOFF5 + SROWS5 * SCOLS5)   /* 5452 */
#define SOFF7 (SOFF6 + SROWS6 * SCOLS6)   /* 5764 */
#define SMEM_FLOATS (SOFF7 + SROWS7 * SCOLS7 + 1) /* 5856 total, 23.4 KB */

__global__ __launch_bounds__(256)
void COOLCHIC_INTERP_ENC_kernel(const f32x2* __restrict__ x,
                                const float* __restrict__ emb,
                                float* __restrict__ out, int n) {
    __shared__ float smem[SMEM_FLOATS];

    // ---- TDM stage: levels 4..7 corner tiles -> LDS (wave 0 issues) ----
    if (threadIdx.x < 32) {
        const int srows[4] = {SROWS4, SROWS5, SROWS6, SROWS7};
        const int scols[4] = {SCOLS4, SCOLS5, SCOLS6, SCOLS7};
        const int soff[4]  = {SOFF4, SOFF5, SOFF6, SOFF7};
#pragma unroll
        for (int s = 0; s < 4; ++s) {
            const int l = 4 + s;
            unsigned long long ga =
                (unsigned long long)(const void*)(emb + (size_t)l * H_GRID * W_GRID);
            // D# group 0: count=1, lds byte addr, 57-bit global addr, type=2
            u32x4 g0;
            g0.x = 1u;
            g0.y = (unsigned)(soff[s] * 4);
            g0.z = (unsigned)ga;
            g0.w = (unsigned)((ga >> 32) & 0x1FFFFFFull) | (2u << 30);
            // D# group 1: data_size=4B (code 2); tensor 1440 x 721, stride 1440;
            // tile = cols x rows; no multicast / iterate / pad / barrier.
            i32x8 g1;
            g1[0] = (int)(2u << 16);                                   // data_size
            g1[1] = (int)((unsigned)W_GRID << 16);                     // dim0[15:0]
            g1[2] = (int)(((unsigned)W_GRID >> 16) |
                          ((unsigned)H_GRID << 16));                   // dim0[31:16] | dim1[15:0]
            g1[3] = (int)(((unsigned)H_GRID >> 16) |
                          ((unsigned)scols[s] << 16));                 // dim1[31:16] | tile_dim0
            g1[4] = srows[s];                                          // tile_dim1 (tile_dim2=0)
            g1[5] = W_GRID;                                            // dim0_stride[31:0]
            g1[6] = 0;
            g1[7] = 0;
            i32x4 gz4 = {0, 0, 0, 0};                                  // 2D: groups 2/3 unused
            i32x8 gz8 = {0, 0, 0, 0, 0, 0, 0, 0};
            __builtin_amdgcn_tensor_load_to_lds(g0, g1, gz4, gz4, gz8, 0);
        }
        __builtin_amdgcn_s_wait_tensorcnt(0);
    }
    __syncthreads();

    // The TDM builtin writes LDS through an address the compiler cannot see
    // (it is an integer field inside the descriptor). Without this escape +
    // memory clobber, LLVM proves smem is never stored and folds the
    // ds_load gathers below to undef (observed: ds=0 in the histogram).
    {
        float* sp = smem;
        asm volatile("" : "+v"(sp)::"memory");
    }

    const int tid    = blockIdx.x * blockDim.x + threadIdx.x;
    const int stride = gridDim.x * blockDim.x;

    for (int i = tid; i < n; i += stride) {
        if (i + stride < n) __builtin_prefetch(&x[i + stride], 0, 1);
        // x is read-once streaming data: non-temporal load keeps L2 free
        // for the embedding stack that every gather depends on.
        const f32x2 p  = __builtin_nontemporal_load(&x[i]);
        const float lat = p.x, lon = p.y;
        float r[NLEVEL];

        // ---- Levels 0..3: gather from global (33 MB stack lives in 192 MB L2) ----
#pragma unroll
        for (int l = 0; l < 4; ++l) {
            const float inv = 4.0f / (float)(1 << l);  // exact: 4,2,1,0.5
            const float la  = (90.0f - lat) * inv;
            const float lo  = lon * inv;
            const int fla = (int)floorf(la);
            const int flo = (int)floorf(lo);
            const int rf = imin(imax(fla, 0), H_GRID - 1);
            const int cf = imin(imax(flo, 0), W_GRID - 1);
            const int rc = imin(imax(fla + 1, 0), H_GRID - 1);
            const int cc = imin(imax(flo + 1, 0), W_GRID - 1);
            const float* __restrict__ g = emb + (size_t)l * H_GRID * W_GRID;
            const float vff = g[rf * W_GRID + cf];
            const float vfc = g[rf * W_GRID + cc];
            const float vcf = g[rc * W_GRID + cf];
            const float vcc = g[rc * W_GRID + cc];
            const float lfr = la - (float)rf;   // clamped-floor fractions (matches ref)
            const float ofr = lo - (float)cf;
            const float vf = vff + ofr * (vfc - vff);
            const float vc = vcf + ofr * (vcc - vcf);
            r[l] = vf + lfr * (vc - vf);
        }

        // ---- Levels 4..7: gather from LDS-staged corner tiles ----
        {
            const int srows[4] = {SROWS4, SROWS5, SROWS6, SROWS7};
            const int scols[4] = {SCOLS4, SCOLS5, SCOLS6, SCOLS7};
            const int soff[4]  = {SOFF4, SOFF5, SOFF6, SOFF7};
#pragma unroll
            for (int s = 0; s < 4; ++s) {
                const int l = 4 + s;
                const float inv = 4.0f / (float)(1 << l);  // 0.25 .. 0.03125
                const float la  = (90.0f - lat) * inv;
                const float lo  = lon * inv;
                const int fla = (int)floorf(la);
                const int flo = (int)floorf(lo);
                const int rows = srows[s], cols = scols[s];
                // Natural index range at these levels never reaches H-1/W-1,
                // so clamping to the tile bounds is exact (and LDS-safe).
                const int rf = imin(imax(fla, 0), rows - 1);
                const int cf = imin(imax(flo, 0), cols - 1);
                const int rc = imin(imax(fla + 1, 0), rows - 1);
                const int cc = imin(imax(flo + 1, 0), cols - 1);
                const int base = soff[s];
                const float vff = smem[base + rf * cols + cf];
                const float vfc = smem[base + rf * cols + cc];
                const float vcf = smem[base + rc * cols + cf];
                const float vcc = smem[base + rc * cols + cc];
                const float lfr = la - (float)rf;
                const float ofr = lo - (float)cf;
                const float vf = vff + ofr * (vfc - vff);
                const float vc = vcf + ofr * (vcc - vcf);
                r[l] = vf + lfr * (vc - vf);
            }
        }

        // ---- Coalesced 2x b128 non-temporal store: out[i*8 + l] ----
        // out is 128 MB write-once data; NT hint avoids evicting the
        // embedding stack from L2.
        f32x4* o = (f32x4*)(out + (size_t)i * NLEVEL);
        f32x4 lo4 = {r[0], r[1], r[2], r[3]};
        f32x4 hi4 = {r[4], r[5], r[6], r[7]};
        __builtin_nontemporal_store(lo4, &o[0]);
        __builtin_nontemporal_store(hi4, &o[1]);
    }
}

extern "C" void kernel_launch(void* const* d_in, const int* in_sizes, int n_in,
                              void* d_out, int out_size, void* d_ws, size_t ws_size,
                              hipStream_t stream) {
    const f32x2* x   = (const f32x2*)d_in[0];     // [N,2] f32
    const float* emb = (const float*)d_in[1];     // [8,721,1440] f32
    float*       out = (float*)d_out;             // [N,8] f32
    const int n = in_sizes[0] / 2;

    const int threads = 256;                       // 8 wave32s per block
    const int blocks  = 2048;                      // persistent; grid-stride loop
    COOLCHIC_INTERP_ENC_kernel<<<blocks, threads, 0, stream>>>(x, emb, out, n);
}